// RidgeAggregator_26766236188974
// MI455X (gfx1250) — compile-verified
//
#include <hip/hip_runtime.h>

typedef __attribute__((ext_vector_type(2))) float v2f;
typedef __attribute__((ext_vector_type(4))) float v4f;
typedef __attribute__((ext_vector_type(8))) float v8f;

#define RK_LAMBDA 0.1f
#define RK_R 32
#define RK_D 512
#define RK_F 1024

__launch_bounds__(256)
__global__ void ridge_agg_kernel(const float* __restrict__ q,
                                 const float* __restrict__ k,
                                 const float* __restrict__ v,
                                 float* __restrict__ out)
{
    __shared__ float k_s[RK_R * RK_D];      // 64 KB: k_b staged once from HBM
    __shared__ float q_s[RK_D];             // 2 KB
    __shared__ float Apart[2][RK_R * RK_R]; // 8 KB: per-K-half Gram partials
    __shared__ float Aug[RK_R * 33];        // augmented [A + lambda*I | rhs], stride 33
    __shared__ float rhsp[8][RK_R];         // rhs chunk partials
    __shared__ float w_s[RK_R];             // solved + normalized weights

    const int b    = blockIdx.x;
    const int tid  = threadIdx.x;
    const int lane = tid & 31;
    const int wave = tid >> 5;

    // ---------- Prefetch v_b (128 KB) into L2 so phase 3 isn't exposed to
    // ---------- full HBM latency behind the Gram + solve phases.
    {
        const char* vb = (const char*)(v + (size_t)b * RK_R * RK_F);
        #pragma unroll
        for (int i = 0; i < 4; ++i) {
            // 256 threads x 4 lines x 128 B = 128 KB coverage
            __builtin_prefetch(vb + ((size_t)(tid + i * 256) * 128), 0, 1);
        }
    }

    // ---------- Phase 0: stage k_b (32x512) and q_b (512) into LDS ----------
    {
        const v4f* kg  = (const v4f*)(k + (size_t)b * RK_R * RK_D);
        v4f*       ks4 = (v4f*)k_s;
        #pragma unroll
        for (int i = 0; i < 16; ++i) {
            int idx = tid + i * 256;        // 4096 float4 total
            ks4[idx] = __builtin_nontemporal_load(&kg[idx]);   // read-once stream
        }
        if (tid < 128) {
            ((v4f*)q_s)[tid] =
                __builtin_nontemporal_load(&((const v4f*)(q + (size_t)b * RK_D))[tid]);
        }
    }
    __syncthreads();

    // ---------- Phase 1a: Gram A = k k^T via V_WMMA_F32_16X16X4_F32 ----------
    // 8 waves: tile = wave%4 -> (rA,rB) in {0,16}^2 ; khalf = wave/4 covers 256 depths.
    {
        const int tile  = wave & 3;
        const int khalf = wave >> 2;
        const int rA    = (tile >> 1) * 16;
        const int rB    = (tile & 1) * 16;
        const int mrow  = lane & 15;        // M (and N) index for this lane
        const int koff  = (lane >> 4) * 2;  // lane-group K pair: {0,1} or {2,3}

        v8f acc = {};
#if __has_builtin(__builtin_amdgcn_wmma_f32_16x16x4_f32)
        const float* arow = &k_s[(rA + mrow) * RK_D];
        const float* brow = &k_s[(rB + mrow) * RK_D];
        #pragma unroll 4
        for (int s = 0; s < 64; ++s) {
            int d0 = khalf * 256 + s * 4 + koff;
            v2f av = *(const v2f*)(arow + d0);   // A[M][d0], A[M][d0+1]
            v2f bv = *(const v2f*)(brow + d0);   // B[d0][N], B[d0+1][N] (= k[N][d0..])
            acc = __builtin_amdgcn_wmma_f32_16x16x4_f32(
                false, av, false, bv, (short)0, acc, false, false);
        }
#else
        // Scalar fallback with identical output mapping (keeps compile alive).
        #pragma unroll
        for (int g = 0; g < 8; ++g) {
            const float* am = &k_s[(rA + g + 8 * (lane >> 4)) * RK_D + khalf * 256];
            const float* bn = &k_s[(rB + (lane & 15)) * RK_D + khalf * 256];
            float s = 0.0f;
            for (int d = 0; d < 256; ++d) s += am[d] * bn[d];
            acc[g] = s;
        }
#endif
        // C/D layout: VGPR g -> M = g + 8*(lane/16), N = lane%16
        float* part = &Apart[khalf][0];
        #pragma unroll
        for (int g = 0; g < 8; ++g) {
            int m = rA + g + 8 * (lane >> 4);
            int n = rB + (lane & 15);
            part[m * RK_R + n] = acc[g];
        }
    }

    // ---------- Phase 1b: rhs = k_b q_b, chunked partial dots ----------
    {
        int r = tid & 31;
        int c = tid >> 5;                    // chunk 0..7 (64 depths each)
        const float* krow = &k_s[r * RK_D + c * 64];
        const float* qp   = &q_s[c * 64];
        float s = 0.0f;
        #pragma unroll 8
        for (int d = 0; d < 64; ++d) s += krow[d] * qp[d];
        rhsp[c][r] = s;
    }
    __syncthreads();

    // ---------- Combine: Aug = [A + lambda*I | rhs] ----------
    for (int idx = tid; idx < RK_R * RK_R; idx += 256) {
        int r = idx >> 5, s = idx & 31;
        float a = Apart[0][idx] + Apart[1][idx];
        if (r == s) a += RK_LAMBDA;
        Aug[r * 33 + s] = a;
    }
    if (tid < RK_R) {
        float s = 0.0f;
        #pragma unroll
        for (int c = 0; c < 8; ++c) s += rhsp[c][tid];
        Aug[tid * 33 + 32] = s;
    }
    __syncthreads();

    // ---------- Phase 2: solve (A + lambda I) w = rhs ----------
    // SPD system: Gaussian elimination without pivoting; row r owned by thread r.
    for (int i = 0; i < RK_R; ++i) {
        if (tid > i && tid < RK_R) {
            float f = Aug[tid * 33 + i] / Aug[i * 33 + i];
            for (int j = i; j <= RK_R; ++j)
                Aug[tid * 33 + j] -= f * Aug[i * 33 + j];
        }
        __syncthreads();
    }
    if (tid == 0) {
        // back substitution into w_s
        for (int r = RK_R - 1; r >= 0; --r) {
            float s = Aug[r * 33 + 32];
            for (int j = r + 1; j < RK_R; ++j) s -= Aug[r * 33 + j] * w_s[j];
            w_s[r] = s / Aug[r * 33 + r];
        }
        // clip(-0.9, 1.1) -> normalize by (sum + 1e-8) -> clip(-10, 10)
        float sum = 0.0f;
        for (int r = 0; r < RK_R; ++r) {
            float x = fminf(fmaxf(w_s[r], -0.9f), 1.1f);
            w_s[r] = x;
            sum += x;
        }
        float inv = 1.0f / (sum + 1e-8f);
        for (int r = 0; r < RK_R; ++r) {
            float x = w_s[r] * inv;
            w_s[r] = fminf(fmaxf(x, -10.0f), 10.0f);
        }
    }
    __syncthreads();

    // ---------- Phase 3: out_b = w^T v_b  (HBM-streaming, the bandwidth term) ----------
    float wreg[RK_R];
    #pragma unroll
    for (int r = 0; r < RK_R; ++r) wreg[r] = w_s[r];

    const v4f* vg = (const v4f*)(v + (size_t)b * RK_R * RK_F);
    v4f o = {0.0f, 0.0f, 0.0f, 0.0f};
    #pragma unroll
    for (int r = 0; r < RK_R; ++r) {
        v4f val = __builtin_nontemporal_load(&vg[r * (RK_F / 4) + tid]); // coalesced, read-once
        o += wreg[r] * val;
    }
    __builtin_nontemporal_store(o, &((v4f*)(out + (size_t)b * RK_F))[tid]);
}

extern "C" void kernel_launch(void* const* d_in, const int* in_sizes, int n_in,
                              void* d_out, int out_size, void* d_ws, size_t ws_size,
                              hipStream_t stream) {
    const float* q = (const float*)d_in[0];   // (B, 512)
    const float* k = (const float*)d_in[1];   // (B, 32, 512)
    const float* v = (const float*)d_in[2];   // (B, 32, 1024)
    float* out = (float*)d_out;               // (B, 1024)
    const int B = in_sizes[0] / RK_D;         // 4096
    ridge_agg_kernel<<<B, 256, 0, stream>>>(q, k, v, out);
}